// DPMamba_50757923504480
// MI455X (gfx1250) — compile-verified
//
#include <hip/hip_runtime.h>
#include <hip/hip_bf16.h>

// ---------------------------------------------------------------------------
// DPMamba forward for MI455X (gfx1250, wave32).
// All GEMMs run through V_WMMA_F32_16X16X4_F32 (full fp32 accuracy).
// Selective scan: 1 block / sequence, h-state in registers, B/C via LDS.
// Whole model (~45MB fp32) lives in the 192MB L2; latency-bound scans
// dominate, so precision-preserving f32 WMMA is the right matrix path.
// ---------------------------------------------------------------------------

#define NCH      64
#define DINNER   128
#define DSTATE   16
#define DTRANK   4
#define KCH      250
#define PCH      125
#define NLAYERS  8

typedef __attribute__((ext_vector_type(2))) float v2f;
typedef __attribute__((ext_vector_type(8))) float v8f;

static inline int ceil_div(int a, int b) { return (a + b - 1) / b; }

// ---------------------------------------------------------------------------
// Generic fp32 WMMA GEMM:  C[M,N] = A[M,K] * B[K,N]
//   BT == 0 : B row-major K x N     (element (k,n) at Bm[k*ldb + n])
//   BT == 1 : B = W^T where W is N x K row-major (element (k,n) at Bm[n*ldb+k])
// One wave (32 lanes) per 16x16 output tile, K stepped by 4.
// A 16x4 layout : lanes0-15 {V0:K=0,V1:K=1}, lanes16-31 {V0:K=2,V1:K=3}, M=lane&15
// B 4x16 layout : mirrored (N=lane&15, K pairs split across lane halves)
// C 16x16 layout: VGPR r -> M = r + 8*(lane>>4), N = lane&15
// ---------------------------------------------------------------------------
template <int BT>
__global__ void gemm_wmma_f32(const float* __restrict__ A,
                              const float* __restrict__ Bm,
                              float* __restrict__ C,
                              int M, int Nc, int Kc,
                              int lda, int ldb, int ldc)
{
    const int mt   = blockIdx.x;
    const int nt   = blockIdx.y;
    const int lane = threadIdx.x;
    const int half = lane >> 4;
    const int l16  = lane & 15;

    int row = mt * 16 + l16; if (row >= M)  row = M - 1;   // clamp loads; stores guarded
    int col = nt * 16 + l16; if (col >= Nc) col = Nc - 1;

    v8f acc = {};
    for (int k = 0; k < Kc; k += 4) {
        const int kb = k + 2 * half;
        v2f a, b;
        a.x = A[(long)row * lda + kb];
        a.y = A[(long)row * lda + kb + 1];
        if (BT) {
            b.x = Bm[(long)col * ldb + kb];
            b.y = Bm[(long)col * ldb + kb + 1];
        } else {
            b.x = Bm[(long)kb * ldb + col];
            b.y = Bm[(long)(kb + 1) * ldb + col];
        }
        acc = __builtin_amdgcn_wmma_f32_16x16x4_f32(
                  false, a, false, b, (short)0, acc, false, false);
    }

    const int cn = nt * 16 + l16;
    if (cn < Nc) {
#pragma unroll
        for (int r = 0; r < 8; ++r) {
            const int cm = mt * 16 + r + 8 * half;
            if (cm < M) C[(long)cm * ldc + cn] = acc[r];
        }
    }
}

// ---------------------------------------------------------------------------
// Encoder: mw[c,t] = relu( sum_k enc_w[c,k] * mix[t*8+k] )
// ---------------------------------------------------------------------------
__global__ void enc_kernel(const float* __restrict__ mix,
                           const float* __restrict__ ew,
                           float* __restrict__ mw, int L)
{
    int i = blockIdx.x * blockDim.x + threadIdx.x;
    if (i >= NCH * L) return;
    int c = i / L, t = i % L;
    float s = 0.f;
    const int base = t * 8;
#pragma unroll
    for (int k = 0; k < 16; ++k) s += ew[c * 16 + k] * mix[base + k];
    mw[i] = s > 0.f ? s : 0.f;
}

// ---------------------------------------------------------------------------
// Global layer norm (gnorm): mean/var over whole tensor, per-channel affine
// ---------------------------------------------------------------------------
__global__ void reduce_meanvar(const float* __restrict__ x, int n,
                               float* __restrict__ stats)
{
    __shared__ float ssum[256], ssq[256];
    float s = 0.f, q = 0.f;
    for (int i = threadIdx.x; i < n; i += 256) { float v = x[i]; s += v; q += v * v; }
    ssum[threadIdx.x] = s; ssq[threadIdx.x] = q;
    __syncthreads();
    for (int st = 128; st > 0; st >>= 1) {
        if ((int)threadIdx.x < st) {
            ssum[threadIdx.x] += ssum[threadIdx.x + st];
            ssq[threadIdx.x]  += ssq[threadIdx.x + st];
        }
        __syncthreads();
    }
    if (threadIdx.x == 0) {
        float m = ssum[0] / (float)n;
        float v = ssq[0] / (float)n - m * m;
        stats[0] = m;
        stats[1] = rsqrtf(v + 1e-8f);
    }
}

__global__ void gnorm_apply(const float* __restrict__ x, float* __restrict__ y,
                            const float* __restrict__ stats,
                            const float* __restrict__ w,
                            const float* __restrict__ b,
                            int chanStride, int n)
{
    int i = blockIdx.x * blockDim.x + threadIdx.x;
    if (i >= n) return;
    int c = i / chanStride;
    y[i] = (x[i] - stats[0]) * stats[1] * w[c] + b[c];
}

// ---------------------------------------------------------------------------
// Segment: x(64,L) -> seg(64,K,S) with 50% overlap; layout idx=(c*K+k)*S+s
// ---------------------------------------------------------------------------
__global__ void segment_kernel(const float* __restrict__ x,
                               float* __restrict__ seg, int L, int S)
{
    int i = blockIdx.x * blockDim.x + threadIdx.x;
    int total = NCH * KCH * S;
    if (i >= total) return;
    int s = i % S, k = (i / S) % KCH, c = i / (S * KCH);
    int ii = s >> 1;
    long pos = (long)ii * KCH + k + ((s & 1) ? 0 : -PCH);
    float v = 0.f;
    if (pos >= 0 && pos < L) v = x[(long)c * L + pos];
    seg[i] = v;
}

// seg(c,k,s) <-> intra tokens (s,k,c) and inter tokens (k,s,c)
__global__ void seg_to_intra(const float* __restrict__ seg, float* __restrict__ x, int S)
{
    int i = blockIdx.x * blockDim.x + threadIdx.x;
    int total = NCH * KCH * S;  if (i >= total) return;
    int s = i % S, k = (i / S) % KCH, c = i / (S * KCH);
    x[((long)s * KCH + k) * NCH + c] = seg[i];
}
__global__ void intra_to_seg(const float* __restrict__ x, float* __restrict__ seg, int S)
{
    int i = blockIdx.x * blockDim.x + threadIdx.x;
    int total = NCH * KCH * S;  if (i >= total) return;
    int s = i % S, k = (i / S) % KCH, c = i / (S * KCH);
    seg[i] = x[((long)s * KCH + k) * NCH + c];
}
__global__ void seg_to_inter(const float* __restrict__ seg, float* __restrict__ x, int S)
{
    int i = blockIdx.x * blockDim.x + threadIdx.x;
    int total = NCH * KCH * S;  if (i >= total) return;
    int s = i % S, k = (i / S) % KCH, c = i / (S * KCH);
    x[((long)k * S + s) * NCH + c] = seg[i];
}
__global__ void inter_to_seg(const float* __restrict__ x, float* __restrict__ seg, int S)
{
    int i = blockIdx.x * blockDim.x + threadIdx.x;
    int total = NCH * KCH * S;  if (i >= total) return;
    int s = i % S, k = (i / S) % KCH, c = i / (S * KCH);
    seg[i] = x[((long)k * S + s) * NCH + c];
}

// ---------------------------------------------------------------------------
// rmsnorm per token (64-dim); safe in-place
// ---------------------------------------------------------------------------
__global__ void rmsnorm_kernel(const float* __restrict__ x,
                               const float* __restrict__ w,
                               float* __restrict__ y, int tokens)
{
    int t = blockIdx.x * blockDim.x + threadIdx.x;
    if (t >= tokens) return;
    const float* xp = x + (long)t * NCH;
    float s = 0.f;
#pragma unroll
    for (int c = 0; c < NCH; ++c) { float v = xp[c]; s += v * v; }
    float r = rsqrtf(s / (float)NCH + 1e-5f);
    float* yp = y + (long)t * NCH;
#pragma unroll
    for (int c = 0; c < NCH; ++c) yp[c] = xp[c] * r * w[c];
}

// ---------------------------------------------------------------------------
// Depthwise causal conv (D_CONV=4) + SiLU. Input = xc half of xz (stride 256).
// rev=1 processes the sequence time-reversed (flip/conv/flip folded in).
// ---------------------------------------------------------------------------
__global__ void mamba_conv(const float* __restrict__ xz,
                           const float* __restrict__ w,
                           const float* __restrict__ b,
                           float* __restrict__ xc,
                           int nseq, int Ls, int rev)
{
    int i = blockIdx.x * blockDim.x + threadIdx.x;
    int total = nseq * Ls * DINNER;
    if (i >= total) return;
    int d = i % DINNER, t = (i / DINNER) % Ls, s = i / (DINNER * Ls);
    float acc = b[d];
#pragma unroll
    for (int j = 0; j < 4; ++j) {
        int tt = rev ? (t + 3 - j) : (t - 3 + j);
        if (tt >= 0 && tt < Ls)
            acc += w[d * 4 + j] * xz[((long)(s * Ls + tt)) * 256 + d];
    }
    xc[i] = acc / (1.f + __expf(-acc));
}

// delta = softplus(dt @ dt_w^T + dt_b), dt = dbl[:, :4]
__global__ void delta_kernel(const float* __restrict__ dbl,
                             const float* __restrict__ dtw,
                             const float* __restrict__ dtb,
                             float* __restrict__ delta, int tokens)
{
    int i = blockIdx.x * blockDim.x + threadIdx.x;
    if (i >= tokens * DINNER) return;
    int d = i % DINNER, t = i / DINNER;
    float s = dtb[d];
#pragma unroll
    for (int r = 0; r < DTRANK; ++r) s += dbl[(long)t * 36 + r] * dtw[d * 4 + r];
    delta[i] = (s > 20.f) ? s : log1pf(__expf(s));
}

// ---------------------------------------------------------------------------
// Selective scan: 1 block (128 threads) per sequence; thread owns channel d,
// h[d,0:16] in registers, A=-exp(A_log) in registers, B/C staged via LDS.
// ---------------------------------------------------------------------------
__global__ void scan_kernel(const float* __restrict__ delta,
                            const float* __restrict__ xc,
                            const float* __restrict__ dbl,
                            const float* __restrict__ A_log,
                            float* __restrict__ y, int Ls, int rev)
{
    const int seq = blockIdx.x;
    const int d   = threadIdx.x;      // 0..127
    float A[DSTATE], h[DSTATE];
#pragma unroll
    for (int s = 0; s < DSTATE; ++s) {
        A[s] = -__expf(A_log[d * DSTATE + s]);
        h[s] = 0.f;
    }
    __shared__ float Bs[DSTATE], Cs[DSTATE];
    for (int step = 0; step < Ls; ++step) {
        const int t = rev ? (Ls - 1 - step) : step;
        const long tok = (long)seq * Ls + t;
        __syncthreads();
        if (d < 16)            Bs[d]      = dbl[tok * 36 + 4 + d];
        else if (d < 32)       Cs[d - 16] = dbl[tok * 36 + 20 + (d - 16)];
        __syncthreads();
        const float dt = delta[tok * DINNER + d];
        const float du = dt * xc[tok * DINNER + d];
        float yv = 0.f;
#pragma unroll
        for (int s = 0; s < DSTATE; ++s) {
            float e = __expf(dt * A[s]);
            h[s] = e * h[s] + du * Bs[s];
            yv  += h[s] * Cs[s];
        }
        y[tok * DINNER + d] = yv;
    }
}

// y = (y + xc*D) * silu(z)   (z = second half of xz)
__global__ void gate_kernel(float* __restrict__ y,
                            const float* __restrict__ xc,
                            const float* __restrict__ xz,
                            const float* __restrict__ Dp, int tokens)
{
    int i = blockIdx.x * blockDim.x + threadIdx.x;
    if (i >= tokens * DINNER) return;
    int d = i % DINNER, t = i / DINNER;
    float v = y[i] + xc[i] * Dp[d];
    float z = xz[(long)t * 256 + DINNER + d];
    y[i] = v * (z / (1.f + __expf(-z)));
}

// elementwise helpers
__global__ void add_inplace(float* __restrict__ x, const float* __restrict__ a, int n)
{
    int i = blockIdx.x * blockDim.x + threadIdx.x;
    if (i < n) x[i] += a[i];
}
__global__ void add2(const float* __restrict__ a, const float* __restrict__ b,
                     float* __restrict__ c, int n)
{
    int i = blockIdx.x * blockDim.x + threadIdx.x;
    if (i < n) c[i] = a[i] + b[i];
}
__global__ void prelu_kernel(float* __restrict__ x, const float* __restrict__ a, int n)
{
    int i = blockIdx.x * blockDim.x + threadIdx.x;
    if (i >= n) return;
    float v = x[i];
    x[i] = v > 0.f ? v : a[0] * v;
}
__global__ void bias_chan_kernel(float* __restrict__ x, const float* __restrict__ b,
                                 int chanStride, int n)
{
    int i = blockIdx.x * blockDim.x + threadIdx.x;
    if (i < n) x[i] += b[i / chanStride];
}

// over_add: m(64,K,S) -> out(64,L)
__global__ void over_add_kernel(const float* __restrict__ m, float* __restrict__ out,
                                int L, int S)
{
    int i = blockIdx.x * blockDim.x + threadIdx.x;
    if (i >= NCH * L) return;
    int c = i / L, j = i % L;
    int p1 = j + PCH;
    int i1 = p1 / KCH, k1 = p1 % KCH;
    int i2 = j / KCH,  k2 = j % KCH;
    float v = 0.f;
    int s1 = 2 * i1;     if (s1 < S) v += m[((long)c * KCH + k1) * S + s1];
    int s2 = 2 * i2 + 1; if (s2 < S) v += m[((long)c * KCH + k2) * S + s2];
    out[i] = v;
}

// mm = tanh(mo + ob[c]) * sigmoid(mg + gb[c])
__global__ void mask_gate_kernel(const float* __restrict__ mo,
                                 const float* __restrict__ mg,
                                 const float* __restrict__ ob,
                                 const float* __restrict__ gb,
                                 float* __restrict__ mm, int L, int n)
{
    int i = blockIdx.x * blockDim.x + threadIdx.x;
    if (i >= n) return;
    int c = i / L;
    float g = mg[i] + gb[c];
    mm[i] = tanhf(mo[i] + ob[c]) * (1.f / (1.f + __expf(-g)));
}

// masked = mw * relu(me)
__global__ void masked_kernel(const float* __restrict__ me,
                              const float* __restrict__ mw,
                              float* __restrict__ masked, int n)
{
    int i = blockIdx.x * blockDim.x + threadIdx.x;
    if (i >= n) return;
    float v = me[i];
    masked[i] = mw[i] * (v > 0.f ? v : 0.f);
}

// Transposed-conv decoder: dec[o] = sum_{c,k:(o+k-15)%8==0} dec_w[c,15-k]*masked[c,(o+k-15)/8]
__global__ void dec_kernel(const float* __restrict__ masked,
                           const float* __restrict__ dw,
                           float* __restrict__ out, int L, int T)
{
    int o = blockIdx.x * blockDim.x + threadIdx.x;
    if (o >= T) return;
    float s = 0.f;
#pragma unroll
    for (int k = 0; k < 16; ++k) {
        int p = o + k - 15;
        if (p >= 0 && (p & 7) == 0) {
            int t = p >> 3;
            if (t < L) {
                for (int c = 0; c < NCH; ++c)
                    s += dw[c * 16 + (15 - k)] * masked[(long)c * L + t];
            }
        }
    }
    out[o] = s;
}

// ---------------------------------------------------------------------------
// Host-side orchestration
// ---------------------------------------------------------------------------
struct MixerP {
    const float *in_w, *conv_w, *conv_b, *x_w, *dt_w, *dt_b, *A_log, *Dp, *out_w;
};
struct Bufs {
    float *h, *mixout, *xz, *xc, *dbl, *delta, *y;
};

static void run_mixer(hipStream_t st, const MixerP& mx, const float* h,
                      float* mixout, int nseq, int Ls, int rev, Bufs& B)
{
    const int tokens = nseq * Ls;
    const int tot128 = tokens * DINNER;
    // in_proj: tokens x 256 = h(tokens x 64) * in_w^T
    gemm_wmma_f32<1><<<dim3(ceil_div(tokens, 16), 16), 32, 0, st>>>(
        h, mx.in_w, B.xz, tokens, 256, 64, 64, 64, 256);
    // depthwise conv + silu
    mamba_conv<<<ceil_div(tot128, 256), 256, 0, st>>>(
        B.xz, mx.conv_w, mx.conv_b, B.xc, nseq, Ls, rev);
    // x_proj: tokens x 36
    gemm_wmma_f32<1><<<dim3(ceil_div(tokens, 16), 3), 32, 0, st>>>(
        B.xc, mx.x_w, B.dbl, tokens, 36, 128, 128, 128, 36);
    // delta
    delta_kernel<<<ceil_div(tot128, 256), 256, 0, st>>>(
        B.dbl, mx.dt_w, mx.dt_b, B.delta, tokens);
    // selective scan
    scan_kernel<<<nseq, DINNER, 0, st>>>(B.delta, B.xc, B.dbl, mx.A_log, B.y, Ls, rev);
    // gate with silu(z) and D skip
    gate_kernel<<<ceil_div(tot128, 256), 256, 0, st>>>(B.y, B.xc, B.xz, mx.Dp, tokens);
    // out_proj: tokens x 64
    gemm_wmma_f32<1><<<dim3(ceil_div(tokens, 16), 4), 32, 0, st>>>(
        B.y, mx.out_w, mixout, tokens, 64, 128, 128, 128, 64);
}

static MixerP make_mixer(void* const* d_in, int b)
{
    auto F = [&](int i) { return (const float*)d_in[i]; };
    // mixer dict insertion order:
    // in_proj_w, conv_w, conv_b, x_proj_w, dt_proj_w, dt_proj_b, A_log, D, out_proj_w
    MixerP m;
    m.in_w = F(b + 0); m.conv_w = F(b + 1); m.conv_b = F(b + 2);
    m.x_w = F(b + 3);  m.dt_w = F(b + 4);   m.dt_b = F(b + 5);
    m.A_log = F(b + 6); m.Dp = F(b + 7);    m.out_w = F(b + 8);
    return m;
}

static void run_bimamba(hipStream_t st, void* const* d_in, int base,
                        float* x, int nseq, int Ls, Bufs& B)
{
    auto F = [&](int i) { return (const float*)d_in[i]; };
    // seq insertion order: blocks[0]{norm_w, fwd(9), bwd(9)}, norm_f
    const int tokens = nseq * Ls;
    const int n64 = tokens * NCH;
    rmsnorm_kernel<<<ceil_div(tokens, 256), 256, 0, st>>>(x, F(base + 0), B.h, tokens);
    MixerP fwd = make_mixer(d_in, base + 1);
    MixerP bwd = make_mixer(d_in, base + 10);
    run_mixer(st, fwd, B.h, B.mixout, nseq, Ls, 0, B);
    add_inplace<<<ceil_div(n64, 256), 256, 0, st>>>(x, B.mixout, n64);
    run_mixer(st, bwd, B.h, B.mixout, nseq, Ls, 1, B);
    add_inplace<<<ceil_div(n64, 256), 256, 0, st>>>(x, B.mixout, n64);
    rmsnorm_kernel<<<ceil_div(tokens, 256), 256, 0, st>>>(x, F(base + 19), x, tokens);
}

extern "C" void kernel_launch(void* const* d_in, const int* in_sizes, int n_in,
                              void* d_out, int out_size, void* d_ws, size_t ws_size,
                              hipStream_t stream)
{
    auto F = [&](int i) { return (const float*)d_in[i]; };

    // ---- input indices (setup_inputs dict insertion order) ----
    // 0: mixture
    // params: enc_w=1, ln_w=2, ln_b=3, bottleneck_w=4, layers=5..356 (8 x 44),
    //         prelu_a=357, conv2d_w=358, conv2d_b=359, out_w=360, out_b=361,
    //         gate_w=362, gate_b=363, end_w=364, dec_w=365
    // per-layer (44): intra-seq(20) [norm_w, fwd(9), bwd(9), norm_f],
    //                 inter-seq(20), intra_norm_w, intra_norm_b,
    //                 inter_norm_w, inter_norm_b
    const int I_ENC = 1, I_LNW = 2, I_LNB = 3, I_BOT = 4, I_LAY = 5;
    const int I_PRELU = 357, I_C2W = 358, I_C2B = 359, I_OW = 360, I_OB = 361;
    const int I_GW = 362, I_GB = 363, I_EW = 364, I_DEC = 365;
    (void)n_in; (void)ws_size;

    const int T = in_sizes[0];                 // 32000
    const int L = (T - 16) / 8 + 1;            // 3999
    const int gap  = KCH - (PCH + L % KCH) % KCH;
    const int Lg   = L + gap;
    const int rows = (Lg + PCH) / KCH;
    const int S    = 2 * rows;                 // 34
    const int SEGN = NCH * KCH * S;            // 544000
    const int TOK  = S * KCH;                  // 8500 (both paths)

    // ---- workspace carve-up ----
    float* w = (float*)d_ws;
    size_t off = 0;
    auto take = [&](size_t n) { float* p = w + off; off += (n + 63) & ~(size_t)63; return p; };

    float* mw     = take((size_t)NCH * L);
    float* xbn    = take((size_t)NCH * L);
    float* xbt    = take((size_t)NCH * L);
    float* seg    = take(SEGN);
    float* tmpA   = take(SEGN);
    float* intraN = take(SEGN);
    float* interN = take(SEGN);
    float* seqx   = take((size_t)TOK * NCH);
    float* stats  = take(8);
    Bufs B;
    B.h      = take((size_t)TOK * NCH);
    B.mixout = take((size_t)TOK * NCH);
    B.xz     = take((size_t)TOK * 256);
    B.xc     = take((size_t)TOK * DINNER);
    B.dbl    = take((size_t)TOK * 36);
    B.delta  = take((size_t)TOK * DINNER);
    B.y      = take((size_t)TOK * DINNER);
    float* m2d    = take(SEGN);
    float* oa     = take((size_t)NCH * L);
    float* mo     = take((size_t)NCH * L);
    float* mg     = take((size_t)NCH * L);
    float* mm     = take((size_t)NCH * L);
    float* masked = take((size_t)NCH * L);

    const int nL  = NCH * L;

    // ---- encoder + gnorm + bottleneck ----
    enc_kernel<<<ceil_div(nL, 256), 256, 0, stream>>>(F(0), F(I_ENC), mw, L);
    reduce_meanvar<<<1, 256, 0, stream>>>(mw, nL, stats);
    gnorm_apply<<<ceil_div(nL, 256), 256, 0, stream>>>(mw, xbn, stats, F(I_LNW), F(I_LNB), L, nL);
    gemm_wmma_f32<0><<<dim3(4, ceil_div(L, 16)), 32, 0, stream>>>(
        F(I_BOT), xbn, xbt, NCH, L, NCH, NCH, L, L);

    // ---- segment ----
    segment_kernel<<<ceil_div(SEGN, 256), 256, 0, stream>>>(xbt, seg, L, S);

    // ---- dual-path layers ----
    for (int l = 0; l < NLAYERS; ++l) {
        const int lb = I_LAY + l * 44;
        // intra path: 34 sequences of length 250
        seg_to_intra<<<ceil_div(SEGN, 256), 256, 0, stream>>>(seg, seqx, S);
        run_bimamba(stream, d_in, lb + 0, seqx, S, KCH, B);
        intra_to_seg<<<ceil_div(SEGN, 256), 256, 0, stream>>>(seqx, tmpA, S);
        reduce_meanvar<<<1, 256, 0, stream>>>(tmpA, SEGN, stats);
        gnorm_apply<<<ceil_div(SEGN, 256), 256, 0, stream>>>(
            tmpA, intraN, stats, F(lb + 40), F(lb + 41), KCH * S, SEGN);
        // inter path: 250 sequences of length 34
        seg_to_inter<<<ceil_div(SEGN, 256), 256, 0, stream>>>(intraN, seqx, S);
        run_bimamba(stream, d_in, lb + 20, seqx, KCH, S, B);
        inter_to_seg<<<ceil_div(SEGN, 256), 256, 0, stream>>>(seqx, tmpA, S);
        reduce_meanvar<<<1, 256, 0, stream>>>(tmpA, SEGN, stats);
        gnorm_apply<<<ceil_div(SEGN, 256), 256, 0, stream>>>(
            tmpA, interN, stats, F(lb + 42), F(lb + 43), KCH * S, SEGN);
        add2<<<ceil_div(SEGN, 256), 256, 0, stream>>>(intraN, interN, seg, SEGN);
    }

    // ---- mask head ----
    prelu_kernel<<<ceil_div(SEGN, 256), 256, 0, stream>>>(seg, F(I_PRELU), SEGN);
    gemm_wmma_f32<0><<<dim3(4, ceil_div(KCH * S, 16)), 32, 0, stream>>>(
        F(I_C2W), seg, m2d, NCH, KCH * S, NCH, NCH, KCH * S, KCH * S);
    bias_chan_kernel<<<ceil_div(SEGN, 256), 256, 0, stream>>>(m2d, F(I_C2B), KCH * S, SEGN);
    over_add_kernel<<<ceil_div(nL, 256), 256, 0, stream>>>(m2d, oa, L, S);
    gemm_wmma_f32<0><<<dim3(4, ceil_div(L, 16)), 32, 0, stream>>>(
        F(I_OW), oa, mo, NCH, L, NCH, NCH, L, L);
    gemm_wmma_f32<0><<<dim3(4, ceil_div(L, 16)), 32, 0, stream>>>(
        F(I_GW), oa, mg, NCH, L, NCH, NCH, L, L);
    mask_gate_kernel<<<ceil_div(nL, 256), 256, 0, stream>>>(
        mo, mg, F(I_OB), F(I_GB), mm, L, nL);
    gemm_wmma_f32<0><<<dim3(4, ceil_div(L, 16)), 32, 0, stream>>>(
        F(I_EW), mm, mo, NCH, L, NCH, NCH, L, L);           // reuse mo for end-proj
    masked_kernel<<<ceil_div(nL, 256), 256, 0, stream>>>(mo, mw, masked, nL);

    // ---- decoder (transposed conv, stride 8, kernel 16) ----
    dec_kernel<<<ceil_div(T, 256), 256, 0, stream>>>(
        masked, F(I_DEC), (float*)d_out, L, T);
    (void)out_size;
}